// AttModel_66700842107248
// MI455X (gfx1250) — compile-verified
//
#include <hip/hip_runtime.h>
#include <hip/hip_bf16.h>

#define PI_D 3.14159265358979323846

typedef __attribute__((ext_vector_type(2))) float v2f;
typedef __attribute__((ext_vector_type(8))) float v8f;

// ---------------------------------------------------------------------------
// DCT setup: orthonormal DCT-II rows 0..9 of size 35.  idct = dct^T, so these
// 350 floats serve both the forward DCTs and the final inverse DCT.
// ---------------------------------------------------------------------------
__global__ void dct_setup_kernel(float* __restrict__ dct10) {
    int tid = threadIdx.x;
    if (tid < 350) {
        int d = tid / 35, v = tid % 35;
        double w = (d == 0) ? sqrt(1.0 / 35.0) : sqrt(2.0 / 35.0);
        dct10[tid] = (float)(w * cos(PI_D * (v + 0.5) * (double)d / 35.0));
    }
}

// ---------------------------------------------------------------------------
// Per-batch prep: im2col for conv1 (query/key), DCT of sliding windows
// (src_value), and DCT of gathered tail (dct_in cols 0..9).
// src: (256, 50, 48) row-major [time][feat].
// ---------------------------------------------------------------------------
__global__ __launch_bounds__(256) void prep_kernel(
    const float* __restrict__ src, const float* __restrict__ dct10,
    float* __restrict__ Bq,    // (256, 288, 5)
    float* __restrict__ Bk,    // (256, 288, 20)
    float* __restrict__ sval,  // (256, 16, 480)   f = c*10+d
    float* __restrict__ dctin) // (256, 48, 20)    cols 0..9 here
{
    int b = blockIdx.x;
    int tid = threadIdx.x;
    const float* S = src + (long)b * 50 * 48;

    for (int idx = tid; idx < 288 * 5; idx += blockDim.x) {
        int kk = idx / 5, t = idx - kk * 5;
        int c = kk / 6, j = kk - c * 6;
        Bq[(long)b * 288 * 5 + idx] = S[(40 + t + j) * 48 + c] * 1e-3f;
    }
    for (int idx = tid; idx < 288 * 20; idx += blockDim.x) {
        int kk = idx / 20, t = idx - kk * 20;
        int c = kk / 6, j = kk - c * 6;
        Bk[(long)b * 288 * 20 + idx] = S[(t + j) * 48 + c] * 1e-3f;
    }
    for (int idx = tid; idx < 16 * 480; idx += blockDim.x) {
        int n = idx / 480, f = idx - n * 480;
        int c = f / 10, d = f - c * 10;
        float s = 0.f;
        for (int v = 0; v < 35; ++v) s += dct10[d * 35 + v] * S[(n + v) * 48 + c];
        sval[(long)b * 16 * 480 + idx] = s;
    }
    for (int idx = tid; idx < 48 * 10; idx += blockDim.x) {
        int c = idx / 10, d = idx - c * 10;
        float s = 0.f;
        for (int v = 0; v < 35; ++v) {
            int g = (v < 10) ? (40 + v) : 49;
            s += dct10[d * 35 + v] * S[g * 48 + c];
        }
        dctin[(long)b * 48 * 20 + c * 20 + d] = s;
    }
}

// ---------------------------------------------------------------------------
// Generic batched fp32 WMMA GEMM: C = epilogue(A x B), row-major everywhere.
// Each wave computes NM vertically-stacked 16x16 tiles: NM independent
// accumulator chains interleaved per K-step (hides WMMA RAW latency), one
// shared B fragment per step.  A is addressed as base + integer offset so the
// backend emits global_load_b64 (not flat_load: flat would tie the loop to
// DScnt).  Ragged N handled branchlessly (clamped column + value select), so
// EXEC is all-1s at every WMMA.
// grid.x = mgroups * ntiles (mg = x % mgroups, nt = x / mgroups),
// grid.y = batch; batch i -> (i/batch_inner, i%batch_inner) with independent
// outer/inner strides for A, B, C, res.
// Epilogue: v = acc; +bias[col]; *scale; act(0 none,1 relu,2 tanh); +res.
// Requires: M % (NM*16) == 0 per m-group, K % 4 == 0, lda/kh even (8B align).
// ---------------------------------------------------------------------------
template <int NM>
__global__ __launch_bounds__(32) void wmma_gemm_kernel(
    const float* __restrict__ A, const float* __restrict__ B,
    float* __restrict__ C,
    int N, int K, int lda, int ldb, int ldc,
    int mgroups, int batch_inner,
    long a_so, long a_si, long b_so, long b_si, long c_so, long c_si,
    const float* __restrict__ bias, float scale, int act,
    const float* __restrict__ res, long r_so, long r_si)
{
    int lane = threadIdx.x;
    int mg = blockIdx.x % mgroups;
    int nt = blockIdx.x / mgroups;
    int bi = blockIdx.y;
    int bo = bi / batch_inner, bii = bi - bo * batch_inner;

    const float* Ab = A + (long)bo * a_so + (long)bii * a_si;
    const float* Bb = B + (long)bo * b_so + (long)bii * b_si;
    float*       Cb = C + (long)bo * c_so + (long)bii * c_si;

    int r  = lane & 15;
    int kh = (lane >> 4) * 2;         // K split: vgpr=k&1, lane-half=k>>1
    int ncol = nt * 16 + r;
    bool colok = ncol < N;
    int ncl = colok ? ncol : (N - 1); // clamped, always-valid column

    long aoff[NM];
#pragma unroll
    for (int i = 0; i < NM; ++i)
        aoff[i] = (long)((mg * NM + i) * 16 + r) * lda + kh;
    long boff = (long)kh * ldb + ncl;

    v8f acc[NM];
#pragma unroll
    for (int i = 0; i < NM; ++i) acc[i] = (v8f){};

    for (int k0 = 0; k0 < K; k0 += 4) {
        float t0 = Bb[boff];
        float t1 = Bb[boff + ldb];
        v2f bfr;
        bfr.x = colok ? t0 : 0.f;
        bfr.y = colok ? t1 : 0.f;
#pragma unroll
        for (int i = 0; i < NM; ++i) {
            v2f a = *(const v2f*)(Ab + aoff[i] + k0);   // global_load_b64
            acc[i] = __builtin_amdgcn_wmma_f32_16x16x4_f32(
                false, a, false, bfr, (short)0, acc[i], false, false);
        }
        boff += 4 * (long)ldb;
    }

    int rowoff = (lane >> 4) * 8;
    int col = nt * 16 + (lane & 15);
    if (col < N) {
        float bv = bias ? bias[col] : 0.f;
        const float* rb = res ? res + (long)bo * r_so + (long)bii * r_si : nullptr;
#pragma unroll
        for (int i = 0; i < NM; ++i) {
            for (int ii = 0; ii < 8; ++ii) {
                int row = (mg * NM + i) * 16 + rowoff + ii;
                float v = acc[i][ii] + bv;
                v *= scale;
                if (act == 1) v = v > 0.f ? v : 0.f;
                else if (act == 2) v = tanhf(v);
                if (rb) v += rb[(long)row * ldc + col];
                Cb[(long)row * ldc + col] = v;
            }
        }
    }
}

// ---------------------------------------------------------------------------
// conv2 as WMMA GEMM with on-the-fly im2col of the relu'd conv1 output.
// A = W2[h] (512 x 2560), B[(c*5+j)][t] = Y1[h,b,c,t+j], relu on store.
// 4 M-tiles per wave share each im2col B fragment (amortizes the /5 index
// math, interleaves 4 accumulator chains).  Branchless T_out masking.
// grid.x = 8 (m-groups of 64 rows), grid.y = H*256.
// ---------------------------------------------------------------------------
__global__ __launch_bounds__(32) void wmma_conv2_kernel(
    const float* __restrict__ W2,  // (4, 512, 512, 5)
    const float* __restrict__ Y1,  // (4, 256, 512, L_in)
    float* __restrict__ Y2,        // (4, 256, 512, T_out)
    int L_in, int T_out)
{
    constexpr int NM = 4;
    int lane = threadIdx.x;
    int mg = blockIdx.x;           // 0..7
    int i  = blockIdx.y;           // h*256 + b
    int h  = i >> 8;

    const float* A     = W2 + (long)h * 512 * 2560;
    const float* Ybase = Y1 + (long)i * 512 * L_in;

    int r  = lane & 15;
    int kh = (lane >> 4) * 2;
    bool tok = r < T_out;
    int tc = tok ? r : 0;          // clamped, always-valid column

    long aoff[NM];
#pragma unroll
    for (int m = 0; m < NM; ++m)
        aoff[m] = (long)((mg * NM + m) * 16 + r) * 2560 + kh;

    v8f acc[NM];
#pragma unroll
    for (int m = 0; m < NM; ++m) acc[m] = (v8f){};

    for (int k0 = 0; k0 < 2560; k0 += 4) {
        int k  = k0 + kh;
        int c0 = k / 5,  j0 = k  - c0 * 5;
        int k1 = k + 1;
        int c1 = k1 / 5, j1 = k1 - c1 * 5;
        float t0 = Ybase[c0 * L_in + tc + j0];
        float t1 = Ybase[c1 * L_in + tc + j1];
        v2f bfr;
        bfr.x = tok ? t0 : 0.f;
        bfr.y = tok ? t1 : 0.f;
#pragma unroll
        for (int m = 0; m < NM; ++m) {
            v2f a = *(const v2f*)(A + aoff[m] + k0);    // global_load_b64
            acc[m] = __builtin_amdgcn_wmma_f32_16x16x4_f32(
                false, a, false, bfr, (short)0, acc[m], false, false);
        }
    }

    int rowoff = (lane >> 4) * 8;
    int col = lane & 15;
    if (col < T_out) {
        float* Cb = Y2 + (long)i * 512 * T_out;
#pragma unroll
        for (int m = 0; m < NM; ++m) {
            for (int ii = 0; ii < 8; ++ii) {
                int row = (mg * NM + m) * 16 + rowoff + ii;
                float v = acc[m][ii];
                Cb[(long)row * T_out + col] = v > 0.f ? v : 0.f;
            }
        }
    }
}

// ---------------------------------------------------------------------------
// Attention: score (q.k over d=512), normalize over k (sum-norm, +1e-15),
// dct_att = attw x src_value, head-concat + linear -> dct_in cols 10..19.
// One block per batch element.
// ---------------------------------------------------------------------------
__global__ __launch_bounds__(256) void score_att_kernel(
    const float* __restrict__ Q2,   // (4, 256, 512)
    const float* __restrict__ K2,   // (4, 256, 512, 16)
    const float* __restrict__ sval, // (256, 16, 480)
    const float* __restrict__ lw,   // (48, 192)
    const float* __restrict__ lb,   // (48)
    float* __restrict__ dctin)      // (256, 48, 20): write cols 10..19
{
    __shared__ float attw[4][16];
    __shared__ float datt[4][480];
    int b = blockIdx.x, tid = threadIdx.x;

    if (tid < 64) {
        int h = tid >> 4, k = tid & 15;
        const float* q  = Q2 + ((long)h * 256 + b) * 512;
        const float* kv = K2 + (((long)h * 256 + b) * 512) * 16 + k;
        float s = 0.f;
        for (int d = 0; d < 512; ++d) s += q[d] * kv[d * 16];
        attw[h][k] = s + 1e-15f;
    }
    __syncthreads();
    if (tid < 4) {
        float sum = 0.f;
        for (int k = 0; k < 16; ++k) sum += attw[tid][k];
        float inv = 1.f / sum;
        for (int k = 0; k < 16; ++k) attw[tid][k] *= inv;
    }
    __syncthreads();
    const float* sv = sval + (long)b * 16 * 480;
    for (int idx = tid; idx < 4 * 480; idx += blockDim.x) {
        int h = idx / 480, f = idx - h * 480;
        float s = 0.f;
        for (int k = 0; k < 16; ++k) s += attw[h][k] * sv[k * 480 + f];
        datt[h][f] = s;
    }
    __syncthreads();
    for (int idx = tid; idx < 480; idx += blockDim.x) {
        int n = idx / 10, d = idx - n * 10;
        float s = lb[n];
        const float* lwn = lw + n * 192;
        for (int h = 0; h < 4; ++h)
            for (int c = 0; c < 48; ++c)
                s += lwn[h * 48 + c] * datt[h][c * 10 + d];
        dctin[(long)b * 48 * 20 + n * 20 + 10 + d] = s;
    }
}

// ---------------------------------------------------------------------------
// Final inverse DCT: out[b][v][0][n] = sum_{d<10} dct10[d][v]*dct_out[b][n][d]
// (idct = dct^T for the orthonormal DCT-II).
// ---------------------------------------------------------------------------
__global__ __launch_bounds__(256) void idct_out_kernel(
    const float* __restrict__ dctout,  // (256, 48, 20)
    const float* __restrict__ dct10,
    float* __restrict__ out)           // (256, 35, 1, 48)
{
    int b = blockIdx.x, tid = threadIdx.x;
    const float* D = dctout + (long)b * 48 * 20;
    for (int idx = tid; idx < 35 * 48; idx += blockDim.x) {
        int v = idx / 48, n = idx - v * 48;
        float s = 0.f;
        for (int d = 0; d < 10; ++d) s += dct10[d * 35 + v] * D[n * 20 + d];
        out[(long)b * 35 * 48 + idx] = s;
    }
}

// ---------------------------------------------------------------------------
// Launch sequence.
// ---------------------------------------------------------------------------
extern "C" void kernel_launch(void* const* d_in, const int* in_sizes, int n_in,
                              void* d_out, int out_size, void* d_ws, size_t ws_size,
                              hipStream_t stream)
{
    const float* src  = (const float*)d_in[0];
    const float* q_w1 = (const float*)d_in[1];
    const float* q_w2 = (const float*)d_in[2];
    const float* k_w1 = (const float*)d_in[3];
    const float* k_w2 = (const float*)d_in[4];
    const float* lw   = (const float*)d_in[5];
    const float* lb   = (const float*)d_in[6];
    const float* giw  = (const float*)d_in[7];
    const float* gia  = (const float*)d_in[8];
    const float* gib  = (const float*)d_in[9];
    const float* gbw  = (const float*)d_in[10];
    const float* gba  = (const float*)d_in[11];
    const float* gbb  = (const float*)d_in[12];
    const float* gow  = (const float*)d_in[13];
    const float* goa  = (const float*)d_in[14];
    const float* gob  = (const float*)d_in[15];
    float* out = (float*)d_out;

    // Workspace layout (~182 MB of fp32)
    float* ws = (float*)d_ws;
    size_t off = 0;
    auto alloc = [&](size_t n) { float* p = ws + off; off += (n + 63) & ~(size_t)63; return p; };
    float* dct10 = alloc(350);
    float* Bq    = alloc((size_t)256 * 288 * 5);
    float* Bk    = alloc((size_t)256 * 288 * 20);
    float* sval  = alloc((size_t)256 * 16 * 480);
    float* dctin = alloc((size_t)256 * 48 * 20);
    float* Y1q   = alloc((size_t)4 * 256 * 512 * 5);
    float* Y1k   = alloc((size_t)4 * 256 * 512 * 20);
    float* Q2    = alloc((size_t)4 * 256 * 512);
    float* K2    = alloc((size_t)4 * 256 * 512 * 16);
    float* T     = alloc((size_t)256 * 48 * 512);
    float* y0    = alloc((size_t)256 * 48 * 512);
    float* y1    = alloc((size_t)256 * 48 * 512);
    float* dctout= alloc((size_t)256 * 48 * 20);
    (void)ws_size; (void)in_sizes; (void)n_in; (void)out_size;

    const float BN = 0.999995000037499687f; // 1/sqrt(1+1e-5)

    dct_setup_kernel<<<1, 512, 0, stream>>>(dct10);
    prep_kernel<<<256, 256, 0, stream>>>(src, dct10, Bq, Bk, sval, dctin);

    // conv1 query: Y1q = relu(W1q[h] (512x288) x Bq[b] (288x5)); NM=4, 8 mgroups
    wmma_gemm_kernel<4><<<dim3(8 * 1, 1024), 32, 0, stream>>>(
        q_w1, Bq, Y1q, 5, 288, 288, 5, 5, 8, 256,
        (long)512 * 288, 0, 0, (long)288 * 5, (long)256 * 512 * 5, (long)512 * 5,
        nullptr, 1.f, 1, nullptr, 0, 0);
    // conv1 key: Y1k = relu(W1k[h] (512x288) x Bk[b] (288x20))
    wmma_gemm_kernel<4><<<dim3(8 * 2, 1024), 32, 0, stream>>>(
        k_w1, Bk, Y1k, 20, 288, 288, 20, 20, 8, 256,
        (long)512 * 288, 0, 0, (long)288 * 20, (long)256 * 512 * 20, (long)512 * 20,
        nullptr, 1.f, 1, nullptr, 0, 0);

    // conv2 (K = 2560, on-the-fly im2col), relu
    wmma_conv2_kernel<<<dim3(8, 1024), 32, 0, stream>>>(q_w2, Y1q, Q2, 5, 1);
    wmma_conv2_kernel<<<dim3(8, 1024), 32, 0, stream>>>(k_w2, Y1k, K2, 20, 16);

    // attention + head concat + linear -> dct_in cols 10..19
    score_att_kernel<<<256, 256, 0, stream>>>(Q2, K2, sval, lw, lb, dctin);

    // ---- GCN ----  (M=48 -> NM=3, single m-group)
    // gc_in: T = dct_in (48x20) @ giw (20x512)
    wmma_gemm_kernel<3><<<dim3(32, 256), 32, 0, stream>>>(
        dctin, giw, T, 512, 20, 20, 512, 512, 1, 256,
        0, (long)48 * 20, 0, 0, 0, (long)48 * 512,
        nullptr, 1.f, 0, nullptr, 0, 0);
    // y0 = tanh((gia @ T + gib) * BN)
    wmma_gemm_kernel<3><<<dim3(32, 256), 32, 0, stream>>>(
        gia, T, y0, 512, 48, 48, 512, 512, 1, 256,
        0, 0, 0, (long)48 * 512, 0, (long)48 * 512,
        gib, BN, 2, nullptr, 0, 0);

    for (int s = 0; s < 2; ++s) {
        const float* w0  = gbw + ((size_t)s * 2 + 0) * 512 * 512;
        const float* w1p = gbw + ((size_t)s * 2 + 1) * 512 * 512;
        const float* a0  = gba + ((size_t)s * 2 + 0) * 48 * 48;
        const float* a1  = gba + ((size_t)s * 2 + 1) * 48 * 48;
        const float* b0  = gbb + ((size_t)s * 2 + 0) * 512;
        const float* b1  = gbb + ((size_t)s * 2 + 1) * 512;
        // j=0: T = y0 @ w0 ; y1 = tanh((a0 @ T + b0) * BN)
        wmma_gemm_kernel<3><<<dim3(32, 256), 32, 0, stream>>>(
            y0, w0, T, 512, 512, 512, 512, 512, 1, 256,
            0, (long)48 * 512, 0, 0, 0, (long)48 * 512,
            nullptr, 1.f, 0, nullptr, 0, 0);
        wmma_gemm_kernel<3><<<dim3(32, 256), 32, 0, stream>>>(
            a0, T, y1, 512, 48, 48, 512, 512, 1, 256,
            0, 0, 0, (long)48 * 512, 0, (long)48 * 512,
            b0, BN, 2, nullptr, 0, 0);
        // j=1: T = y1 @ w1 ; y0 = tanh((a1 @ T + b1) * BN) + y0 (residual)
        wmma_gemm_kernel<3><<<dim3(32, 256), 32, 0, stream>>>(
            y1, w1p, T, 512, 512, 512, 512, 512, 1, 256,
            0, (long)48 * 512, 0, 0, 0, (long)48 * 512,
            nullptr, 1.f, 0, nullptr, 0, 0);
        wmma_gemm_kernel<3><<<dim3(32, 256), 32, 0, stream>>>(
            a1, T, y0, 512, 48, 48, 512, 512, 1, 256,
            0, 0, 0, (long)48 * 512, 0, (long)48 * 512,
            b1, BN, 2, y0, 0, (long)48 * 512);
    }

    // gc_out: T20 = y0 (48x512) @ gow (512x20)  (reuse T with ldc=20)
    wmma_gemm_kernel<3><<<dim3(2, 256), 32, 0, stream>>>(
        y0, gow, T, 20, 512, 512, 20, 20, 1, 256,
        0, (long)48 * 512, 0, 0, 0, (long)48 * 20,
        nullptr, 1.f, 0, nullptr, 0, 0);
    // dct_out = goa @ T20 + gob + dct_in
    wmma_gemm_kernel<3><<<dim3(2, 256), 32, 0, stream>>>(
        goa, T, dctout, 20, 48, 48, 20, 20, 1, 256,
        0, 0, 0, (long)48 * 20, 0, (long)48 * 20,
        gob, 1.f, 0, dctin, 0, (long)48 * 20);

    // inverse DCT to output (256, 35, 1, 48)
    idct_out_kernel<<<256, 256, 0, stream>>>(dctout, dct10, out);
}